// Loss_yolov1_38276748542372
// MI455X (gfx1250) — compile-verified
//
#include <hip/hip_runtime.h>
#include <hip/hip_bf16.h>
#include <math.h>

#define SG     7
#define NCH    30
#define BLK    256

typedef __attribute__((ext_vector_type(2))) float v2f;
typedef __attribute__((ext_vector_type(8))) float v8f;

// Wave32 full-lane sum using the CDNA5 matrix pipe.
// A (16x4 f32): lane L<16 holds A[L][0],A[L][1]; lane L>=16 holds A[L-16][2],A[L-16][3].
// Set A[.][0]=v_m (lanes 0..15), A[.][2]=v_{m+16} (lanes 16..31), other K-slot = 0,
// B = all ones (4x16)  =>  D[m][n] = v_m + v_{m+16}  for every n.
// Lane n holds D[r + 8*(n/16)][n%16] in its 8 D VGPRs, so
// sum(own 8 regs) gives a half-sum; adding the lane^16 partner's half-sum gives
// the full 32-lane total (available in all lanes).
__device__ __forceinline__ float wave_sum32(float v) {
  v2f a; a.x = v;    a.y = 0.0f;   // data in K-slot 0 (or 2), zero in K-slot 1 (or 3)
  v2f b; b.x = 1.0f; b.y = 1.0f;   // B = ones
  v8f c = {};
  v8f d = __builtin_amdgcn_wmma_f32_16x16x4_f32(
      /*neg_a=*/false, a, /*neg_b=*/false, b,
      /*c_mod=*/(short)0, c, /*reuse_a=*/false, /*reuse_b=*/false);
  float s = ((d[0] + d[1]) + (d[2] + d[3])) + ((d[4] + d[5]) + (d[6] + d[7]));
  s += __shfl_xor(s, 16, 32);
  return s;
}

__global__ void __launch_bounds__(BLK)
yolo_loss_main(const float* __restrict__ pred, const float* __restrict__ targ,
               float* __restrict__ blk_part, int ncells) {
  const int i = blockIdx.x * BLK + threadIdx.x;

  float l_xywh = 0.0f, l_obj = 0.0f, l_noobj = 0.0f, l_cls = 0.0f;

  if (i < ncells) {
    const float* pc = pred + (size_t)i * NCH;
    const float* tc = targ + (size_t)i * NCH;

    const float inv_s = 1.0f / (float)SG;

    // target box + masks
    const float tx = tc[0], ty = tc[1], tw = tc[2], th = tc[3], tconf = tc[4];
    const float coord = (tconf > 0.0f) ? 1.0f : 0.0f;
    const float noobj = (tconf == 0.0f) ? 1.0f : 0.0f;

    const float tx0 = tx * inv_s - 0.5f * tw, ty0 = ty * inv_s - 0.5f * th;
    const float tx1 = tx * inv_s + 0.5f * tw, ty1 = ty * inv_s + 0.5f * th;
    const float area_t = (tx1 - tx0) * (ty1 - ty0);

    // predicted boxes
    const float p0x = pc[0], p0y = pc[1], p0w = pc[2], p0h = pc[3], p0c = pc[4];
    const float p1x = pc[5], p1y = pc[6], p1w = pc[7], p1h = pc[8], p1c = pc[9];

    auto iou_f = [&](float bx, float by, float bw, float bh) -> float {
      const float x0 = bx * inv_s - 0.5f * bw, y0 = by * inv_s - 0.5f * bh;
      const float x1 = bx * inv_s + 0.5f * bw, y1 = by * inv_s + 0.5f * bh;
      const float ltx = fmaxf(x0, tx0), lty = fmaxf(y0, ty0);
      const float rbx = fminf(x1, tx1), rby = fminf(y1, ty1);
      const float w = fmaxf(rbx - ltx, 0.0f), h = fmaxf(rby - lty, 0.0f);
      const float inter = w * h;
      const float area_p = (x1 - x0) * (y1 - y0);
      return inter / (area_p + area_t - inter);
    };

    const float iou0 = iou_f(p0x, p0y, p0w, p0h);
    const float iou1 = iou_f(p1x, p1y, p1w, p1h);
    const bool  b1   = (iou1 > iou0);            // argmax: tie -> box 0
    const float max_iou = fmaxf(iou0, iou1);

    const float rx = b1 ? p1x : p0x, ry = b1 ? p1y : p0y;
    const float rw = b1 ? p1w : p0w, rh = b1 ? p1h : p0h;
    const float rc = b1 ? p1c : p0c;

    const float dx = rx - tx, dy = ry - ty;
    const float dw = sqrtf(rw) - sqrtf(tw), dh = sqrtf(rh) - sqrtf(th);
    l_xywh = coord * ((dx * dx + dy * dy) + (dw * dw + dh * dh));

    const float dob = rc - max_iou;
    l_obj = coord * dob * dob;

    float cls = 0.0f;
#pragma unroll
    for (int c2 = 10; c2 < 30; ++c2) {
      const float d2 = pc[c2] - tc[c2];
      cls += d2 * d2;
    }
    l_cls = coord * cls;

    const float d4 = p0c - tconf, d9 = p1c - tc[9];
    l_noobj = noobj * (d4 * d4 + d9 * d9);
  }

  // Full-wave WMMA reductions (EXEC all ones here: unconditional code path).
  const float s0 = wave_sum32(l_xywh);
  const float s1 = wave_sum32(l_obj);
  const float s2 = wave_sum32(l_noobj);
  const float s3 = wave_sum32(l_cls);

  __shared__ float lds[BLK / 32][4];
  const int lane = threadIdx.x & 31;
  const int wv   = threadIdx.x >> 5;
  if (lane == 0) {
    lds[wv][0] = s0; lds[wv][1] = s1; lds[wv][2] = s2; lds[wv][3] = s3;
  }
  __syncthreads();
  if (threadIdx.x < 4) {
    float acc = 0.0f;
#pragma unroll
    for (int w2 = 0; w2 < BLK / 32; ++w2) acc += lds[w2][threadIdx.x];
    blk_part[(size_t)blockIdx.x * 4 + threadIdx.x] = acc;
  }
}

__global__ void __launch_bounds__(BLK)
yolo_loss_final(const float* __restrict__ blk_part, int nblk,
                float* __restrict__ out, float inv_bs) {
  float a0 = 0.0f, a1 = 0.0f, a2 = 0.0f, a3 = 0.0f;
  for (int j = threadIdx.x; j < nblk; j += BLK) {
    a0 += blk_part[(size_t)j * 4 + 0];
    a1 += blk_part[(size_t)j * 4 + 1];
    a2 += blk_part[(size_t)j * 4 + 2];
    a3 += blk_part[(size_t)j * 4 + 3];
  }
  const float s0 = wave_sum32(a0);
  const float s1 = wave_sum32(a1);
  const float s2 = wave_sum32(a2);
  const float s3 = wave_sum32(a3);

  __shared__ float lds[BLK / 32][4];
  const int lane = threadIdx.x & 31;
  const int wv   = threadIdx.x >> 5;
  if (lane == 0) {
    lds[wv][0] = s0; lds[wv][1] = s1; lds[wv][2] = s2; lds[wv][3] = s3;
  }
  __syncthreads();
  if (threadIdx.x == 0) {
    float S0 = 0.0f, S1 = 0.0f, S2 = 0.0f, S3 = 0.0f;
#pragma unroll
    for (int w2 = 0; w2 < BLK / 32; ++w2) {
      S0 += lds[w2][0]; S1 += lds[w2][1]; S2 += lds[w2][2]; S3 += lds[w2][3];
    }
    // (loss, loss_xy+loss_wh, loss_obj, loss_noobj, loss_class)
    out[0] = (5.0f * S0 + S1 + 0.5f * S2 + S3) * inv_bs;
    out[1] = S0;
    out[2] = S1;
    out[3] = S2;
    out[4] = S3;
  }
}

extern "C" void kernel_launch(void* const* d_in, const int* in_sizes, int n_in,
                              void* d_out, int out_size, void* d_ws, size_t ws_size,
                              hipStream_t stream) {
  (void)n_in; (void)out_size; (void)ws_size;
  const float* pred = (const float*)d_in[0];
  const float* targ = (const float*)d_in[1];

  const int ncells = in_sizes[0] / NCH;            // 8192*7*7 = 401408
  const int bs     = ncells / (SG * SG);           // 8192
  const int nblk   = (ncells + BLK - 1) / BLK;     // 1568

  float* blk_part = (float*)d_ws;                  // nblk*4 floats of scratch

  yolo_loss_main<<<nblk, BLK, 0, stream>>>(pred, targ, blk_part, ncells);
  yolo_loss_final<<<1, BLK, 0, stream>>>(blk_part, nblk, (float*)d_out,
                                         1.0f / (float)bs);
}